// TorchDispatchModule_79276506349866
// MI455X (gfx1250) — compile-verified
//
#include <hip/hip_runtime.h>
#include <stdint.h>

// ---- problem constants (from reference) ------------------------------------
#define NUM_CHIPS 8
#define NE        64          // N_ROUTED_EXPERTS
#define MAX_TOK   2560
#define SEQ       2048
#define TOPK      8
#define HIDDEN    1024
#define META_LEN  8
#define NPC       (SEQ * TOPK)   // 16384 assignments per chip (2^14)

typedef __attribute__((ext_vector_type(4))) float f32x4;

__device__ __forceinline__ uint64_t rfl64(uint64_t v) {
  uint32_t lo = __builtin_amdgcn_readfirstlane((uint32_t)v);
  uint32_t hi = __builtin_amdgcn_readfirstlane((uint32_t)(v >> 32));
  return ((uint64_t)hi << 32) | lo;
}

// ---- K1: per-chip counting-sort -------------------------------------------
// One block per chip. Thread t owns assignments [t*128, t*128+128) in scan
// order. Pass A: per-thread histogram in LDS (stride 65 to spread banks).
// Pass B: threads 0..63 do an exclusive prefix over thread-histograms for
// their expert (bank-conflict-free: thread e always hits bank (t+e)&63).
// Pass C: rescan, post-incrementing the prefix base -> stable ranks.
#define K1_T 128
#define K1_CHUNK (NPC / K1_T)   // 128

__global__ __launch_bounds__(K1_T) void k1_count_rank(const int* __restrict__ idx,
                                                      int* __restrict__ counts,
                                                      int* __restrict__ ranks) {
  __shared__ int hist[K1_T * 65];   // 33,280 B
  const int c = blockIdx.x;
  const int t = threadIdx.x;
  const int* __restrict__ row = idx + c * NPC;
  int* h = &hist[t * 65];
#pragma unroll
  for (int e = 0; e < NE; ++e) h[e] = 0;
  const int base = t * K1_CHUNK;
  __builtin_prefetch(row + base, 0, 1);   // global_prefetch_b8
  for (int i = 0; i < K1_CHUNK; ++i) h[row[base + i] & 63]++;
  __syncthreads();
  if (t < NE) {
    int run = 0;
    for (int tt = 0; tt < K1_T; ++tt) {
      int v = hist[tt * 65 + t];
      hist[tt * 65 + t] = run;          // exclusive prefix for thread tt
      run += v;
    }
    counts[c * NE + t] = run;           // per-chip, per-expert count
  }
  __syncthreads();
  int* rr = ranks + c * NPC + base;
  for (int i = 0; i < K1_CHUNK; ++i) {
    int e = row[base + i] & 63;
    rr[i] = h[e]++;                     // stable rank within chip
  }
}

// ---- K2: exclusive cumsum over chips + expert totals -----------------------
__global__ void k2_offsets(const int* __restrict__ counts,
                           int* __restrict__ chip_off,
                           int* __restrict__ totals,
                           int* __restrict__ out_tpe) {
  const int e = threadIdx.x;
  if (e >= NE) return;
  int run = 0;
#pragma unroll
  for (int c = 0; c < NUM_CHIPS; ++c) {
    chip_off[c * NE + e] = run;
    run += counts[c * NE + e];
  }
  totals[e]  = run;
  out_tpe[e] = run;   // tokens_per_expert (int32 bits), flat (8,8) order == e
}

// ---- K3: scatter token rows via async LDS-bypass copies --------------------
// 8 waves / block, one wave per assignment. Each wave copies its 4 KB row:
// 8 iterations of 32 lanes x 16 B memory->LDS, wait ASYNCcnt, then 8
// iterations LDS->memory. No VGPR round-trip for the payload.
__global__ __launch_bounds__(256) void k3_scatter(const float* __restrict__ x,
                                                  const float* __restrict__ w,
                                                  const int* __restrict__ idx,
                                                  const int* __restrict__ ranks,
                                                  const int* __restrict__ chip_off,
                                                  float* __restrict__ disp,
                                                  float* __restrict__ meta) {
  __shared__ char lds_raw[8 * 4096];    // one 4 KB staging slab per wave
  const int lane = threadIdx.x & 31;
  const int wid  = threadIdx.x >> 5;
  const int a = blockIdx.x * 8 + wid;   // assignment id, 0..C*NPC-1 (exact)
  const int c = a >> 14;                // a / NPC
  const int n = a & (NPC - 1);
  const int s = n >> 3;
  const int k = n & 7;
  const int e = idx[a] & 63;
  const int r = chip_off[c * NE + e] + ranks[a];
  if (r >= MAX_TOK) return;             // reference's mode="drop"

  const uint64_t src = rfl64((uint64_t)(uintptr_t)(x + ((size_t)c * SEQ + s) * HIDDEN));
  const uint64_t dst = rfl64((uint64_t)(uintptr_t)(disp + ((size_t)e * MAX_TOK + r) * HIDDEN));
  const unsigned lds0 = (unsigned)(uintptr_t)(void*)&lds_raw[wid * 4096] + (unsigned)(lane * 16);
  const unsigned g0   = (unsigned)(lane * 16);

#pragma unroll
  for (int j = 0; j < 8; ++j) {
    asm volatile("global_load_async_to_lds_b128 %0, %1, %2"
                 :: "v"(lds0 + j * 512), "v"(g0 + j * 512), "s"(src)
                 : "memory");
  }
  asm volatile("s_wait_asynccnt 0x0" ::: "memory");   // loads landed in LDS
#pragma unroll
  for (int j = 0; j < 8; ++j) {
    asm volatile("global_store_async_from_lds_b128 %0, %1, %2"
                 :: "v"(g0 + j * 512), "v"(lds0 + j * 512), "s"(dst)
                 : "memory");
  }
  if (lane == 0) {                      // metadata: [c, s, k, e, w, 0, 0, 0]
    f32x4* mp = (f32x4*)(meta + ((size_t)e * MAX_TOK + r) * META_LEN);
    const float wv = w[(c * SEQ + s) * TOPK + k];
    f32x4 m0 = {(float)c, (float)s, (float)k, (float)e};
    f32x4 m1 = {wv, 0.f, 0.f, 0.f};
    mp[0] = m0;
    mp[1] = m1;
  }
  asm volatile("s_wait_asynccnt 0x0" ::: "memory");   // stores retired
}

// ---- K4: zero / -1 fill of unused slots (disjoint from K3's rows) ----------
__global__ __launch_bounds__(256) void k4_fill(const int* __restrict__ totals,
                                               float* __restrict__ disp,
                                               float* __restrict__ meta) {
  const int lane  = threadIdx.x & 31;
  const int wid   = threadIdx.x >> 5;
  const int rowid = blockIdx.x * 8 + wid;   // 0..NE*MAX_TOK-1 (exact)
  const int e = rowid / MAX_TOK;
  const int r = rowid - e * MAX_TOK;
  if (r < totals[e]) return;                // used row: written by K3
  f32x4 z = {0.f, 0.f, 0.f, 0.f};
  f32x4* dp = (f32x4*)(disp + (size_t)rowid * HIDDEN);
#pragma unroll
  for (int j = 0; j < 8; ++j)
    __builtin_nontemporal_store(z, dp + lane + j * 32);   // streamed, NT hint
  if (lane == 0) {
    f32x4 m = {-1.f, -1.f, -1.f, -1.f};
    f32x4* mp = (f32x4*)(meta + (size_t)rowid * META_LEN);
    __builtin_nontemporal_store(m, mp);
    __builtin_nontemporal_store(m, mp + 1);
  }
}

// ---- launch ----------------------------------------------------------------
extern "C" void kernel_launch(void* const* d_in, const int* in_sizes, int n_in,
                              void* d_out, int out_size, void* d_ws, size_t ws_size,
                              hipStream_t stream) {
  const float* x   = (const float*)d_in[0];   // (8, 2048, 1024) f32
  const float* w   = (const float*)d_in[1];   // (8, 2048, 8)    f32
  const int*   idx = (const int*)  d_in[2];   // (8, 2048, 8)    i32

  float* disp = (float*)d_out;                                    // 64*2560*1024
  float* meta = disp + (size_t)NE * MAX_TOK * HIDDEN;             // 64*2560*8
  int*   tpe  = (int*)(meta + (size_t)NE * MAX_TOK * META_LEN);   // 64 (int32 bits)

  // workspace: [counts 512][chip_off 512][totals 64][pad][ranks 131072] ints
  int* ws       = (int*)d_ws;
  int* counts   = ws;
  int* chip_off = ws + 512;
  int* totals   = ws + 1024;
  int* ranks    = ws + 2048;   // ~532 KB total

  k1_count_rank<<<NUM_CHIPS, K1_T, 0, stream>>>(idx, counts, ranks);
  k2_offsets  <<<1, NE, 0, stream>>>(counts, chip_off, totals, tpe);
  k3_scatter  <<<(NUM_CHIPS * NPC) / 8, 256, 0, stream>>>(x, w, idx, ranks,
                                                          chip_off, disp, meta);
  k4_fill     <<<(NE * MAX_TOK) / 8, 256, 0, stream>>>(totals, disp, meta);
}